// TransLayer1_59897613910227
// MI455X (gfx1250) — compile-verified
//
#include <hip/hip_runtime.h>
#include <math.h>

// ---------------------------------------------------------------------------
// MI455X (gfx1250) Nystromformer layer, fp32 WMMA (V_WMMA_F32_16X16X4_F32).
// Bandwidth-bound (256MB attn output) => fp32 WMMA keeps reference precision
// for free. GEMM tiles double-buffered in LDS via GLOBAL_LOAD_ASYNC_TO_LDS
// (ASYNCcnt) so the inner loop is pure ds_load + v_wmma. wave32 throughout.
// ---------------------------------------------------------------------------

typedef float v2f __attribute__((ext_vector_type(2)));
typedef float v8f __attribute__((ext_vector_type(8)));

__device__ __forceinline__ v8f wmma_f32_4(v2f a, v2f b, v8f c) {
  // (neg_a, A, neg_b, B, c_mod, C, reuse_a, reuse_b)
  return __builtin_amdgcn_wmma_f32_16x16x4_f32(false, a, false, b, (short)0, c,
                                               false, false);
}

// Async copy 16B global -> LDS. VDST = LDS byte offset, VADDR = 64-bit addr.
// Generic shared pointers carry the LDS offset in their low 32 bits.
__device__ __forceinline__ void async_copy_b128(const float* gptr,
                                                const float* lptr) {
  unsigned loff = (unsigned)(size_t)lptr;
  unsigned long long g = (unsigned long long)(size_t)gptr;
  asm volatile("global_load_async_to_lds_b128 %0, %1, off"
               :: "v"(loff), "v"(g)
               : "memory");
}

__device__ __forceinline__ void wait_async0() {
#if __has_builtin(__builtin_amdgcn_s_wait_asynccnt)
  __builtin_amdgcn_s_wait_asynccnt(0);
#else
  asm volatile("s_wait_asynccnt 0" ::: "memory");
#endif
}

// A fragment: 16x4 fp32 tile from LDS, leading dim lda (floats).
// Lanes 0-15 & 16-31 hold M=0..15; VGPR0 = K{0,2}, VGPR1 = K{1,3}.
__device__ __forceinline__ v2f afrag(const float* A, int lda, int lane, int k0) {
  int m = lane & 15;
  int kb = k0 + ((lane >> 4) << 1);
  v2f r;
  r.x = A[m * lda + kb];
  r.y = A[m * lda + kb + 1];
  return r;
}

// ---------------------------------------------------------------------------
// Universal batched GEMM: C = alpha*(A @ op(B)) [+ bias[n]] [+ resid[m][n]]
// A: [M,K] lda.  TB=0: B [K,N] ldb.  TB=1: B [N,K] ldb (C = A@B^T).
// Block: 256 thr (8 waves), tile 16(M) x 128(N), K chunks of 64, K%64==0.
// A and B chunks double-buffered in LDS via async copy.
// ---------------------------------------------------------------------------
template <int TB>
__global__ void gemm_kernel(float* __restrict__ C, const float* __restrict__ A,
                            const float* __restrict__ Bm, int M, int N, int K,
                            int lda, int ldb, int ldc, long long sA,
                            long long sB, long long sC, float alpha,
                            const float* __restrict__ bias,
                            const float* __restrict__ resid, int ldres) {
  constexpr int BROWS = TB ? 128 : 64;   // NT: [n][k]   NN: [k][n]
  constexpr int BCOLS = TB ? 68 : 132;   // padded, row stride mult of 16B
  __shared__ float As[2][16 * 68];
  __shared__ float Bs[2][BROWS * BCOLS];

  const int bz = blockIdx.z;
  const float* Ab = A + (size_t)bz * sA;
  const float* Bb = Bm + (size_t)bz * sB;
  float* Cb = C + (size_t)bz * sC;

  const int m0 = blockIdx.x * 16;
  const int n0blk = blockIdx.y * 128;
  const int wave = threadIdx.x >> 5;
  const int lane = threadIdx.x & 31;
  const int wn = wave * 16;
  const int n0 = n0blk + wn;
  const bool active = (n0 < N);
  const int t = threadIdx.x;

  // stage chunk kc into buffer p (async, no wait)
  auto issue = [&](int p, int kc) {
    {  // A: 16 rows x 64 cols = 256 x b128, one per thread
      int r = t >> 4, c4 = (t & 15) * 4;
      async_copy_b128(&Ab[(size_t)(m0 + r) * lda + kc + c4],
                      &As[p][r * 68 + c4]);
    }
    if (TB) {  // B^T tile: 128 n-rows x 64 k-cols
#pragma unroll
      for (int j = 0; j < 8; ++j) {
        int idx = t + j * 256;
        int r = idx >> 4, c4 = (idx & 15) * 4;
        int gn = n0blk + r;
        if (gn < N)
          async_copy_b128(&Bb[(size_t)gn * ldb + kc + c4],
                          &Bs[p][r * BCOLS + c4]);
      }
    } else {  // B tile: 64 k-rows x 128 n-cols
#pragma unroll
      for (int j = 0; j < 8; ++j) {
        int idx = t + j * 256;
        int r = idx >> 5, c4 = (idx & 31) * 4;
        int gn = n0blk + c4;
        if (gn < N)
          async_copy_b128(&Bb[(size_t)(kc + r) * ldb + gn],
                          &Bs[p][r * BCOLS + c4]);
      }
    }
  };

  v8f acc = {};
  int p = 0;
  issue(0, 0);
  wait_async0();
  __syncthreads();
  for (int kc = 0; kc < K; kc += 64) {
    if (kc + 64 < K) issue(p ^ 1, kc + 64);  // overlap fill with compute
    if (active) {
      const float* Ap = &As[p][0];
      const float* Bp = &Bs[p][0];
      const int kl2 = (lane >> 4) << 1;
      const int n = lane & 15;
#pragma unroll
      for (int k0 = 0; k0 < 64; k0 += 4) {
        v2f a = afrag(Ap, 68, lane, k0);
        int kb = k0 + kl2;
        v2f b;
        if (TB) {
          b.x = Bp[(wn + n) * BCOLS + kb];
          b.y = Bp[(wn + n) * BCOLS + kb + 1];
        } else {
          b.x = Bp[kb * BCOLS + wn + n];
          b.y = Bp[(kb + 1) * BCOLS + wn + n];
        }
        acc = wmma_f32_4(a, b, acc);
      }
    }
    wait_async0();
    __syncthreads();
    p ^= 1;
  }

  if (active) {
    const int n = lane & 15;
    const int mh = (lane >> 4) * 8;  // C/D: VGPR r -> M = r + 8*(lane>=16)
#pragma unroll
    for (int r = 0; r < 8; ++r) {
      int m = m0 + mh + r;
      float v = alpha * acc[r];
      if (bias) v += bias[n0 + n];
      if (resid) v += resid[(size_t)m * ldres + n0 + n];
      Cb[(size_t)m * ldc + n0 + n] = v;
    }
  }
}

// ---------------------------------------------------------------------------
// Fused LayerNorm + QKV projection. Block = 128 thr (4 waves) handles 16 rows.
// GEMM 16x192, K=512 from LDS; q scaled by DIM_HEAD^-0.5.
// ---------------------------------------------------------------------------
__global__ void ln_qkv_kernel(const float* __restrict__ x,
                              const float* __restrict__ gamma,
                              const float* __restrict__ beta,
                              const float* __restrict__ wqkv,
                              float* __restrict__ q, float* __restrict__ kout,
                              float* __restrict__ vout) {
  __shared__ float xs[16][513];
  __shared__ float sred[16][8], qred[16][8];
  __shared__ float mu[16], rstd[16];
  const int t = threadIdx.x;
  const size_t row0 = (size_t)blockIdx.x * 16;

  for (int i = t; i < 16 * 512; i += 128) {
    int r = i >> 9, c = i & 511;
    xs[r][c] = x[(row0 + r) * 512 + c];
  }
  __syncthreads();

  const int r = t >> 3, pp = t & 7;
  {
    float s = 0.f, s2 = 0.f;
    const float* xr = &xs[r][pp * 64];
    for (int c = 0; c < 64; ++c) {
      float vv = xr[c];
      s += vv;
      s2 += vv * vv;
    }
    sred[r][pp] = s;
    qred[r][pp] = s2;
  }
  __syncthreads();
  if (t < 16) {
    float a = 0.f, b2 = 0.f;
    for (int j = 0; j < 8; ++j) {
      a += sred[t][j];
      b2 += qred[t][j];
    }
    float m = a * (1.f / 512.f);
    float var = b2 * (1.f / 512.f) - m * m;
    mu[t] = m;
    rstd[t] = rsqrtf(var + 1e-5f);
  }
  __syncthreads();
  {
    float m = mu[r], rs = rstd[r];
    for (int c = 0; c < 64; ++c) {
      int col = pp * 64 + c;
      xs[r][col] = (xs[r][col] - m) * rs * gamma[col] + beta[col];
    }
  }
  __syncthreads();

  const int wave = t >> 5, lane = t & 31;
  for (int ct = wave; ct < 12; ct += 4) {  // 12 col tiles of 16 -> 192
    v8f acc = {};
    const float* Bp = wqkv + ct * 16;
    for (int k0 = 0; k0 < 512; k0 += 4) {
      v2f a = afrag(&xs[0][0], 513, lane, k0);
      int kb = k0 + ((lane >> 4) << 1);
      int n = lane & 15;
      v2f b;
      b.x = Bp[(size_t)kb * 192 + n];
      b.y = Bp[(size_t)(kb + 1) * 192 + n];
      acc = wmma_f32_4(a, b, acc);
    }
    const int n = lane & 15, mh = (lane >> 4) * 8;
#pragma unroll
    for (int rr = 0; rr < 8; ++rr) {
      size_t grow = row0 + mh + rr;
      int col = ct * 16 + n;
      float val = acc[rr];
      if (col < 64)
        q[grow * 64 + col] = val * 0.125f;  // SCALE = 64^-0.5
      else if (col < 128)
        kout[grow * 64 + (col - 64)] = val;
      else
        vout[grow * 64 + (col - 128)] = val;
    }
  }
}

// landmarks: mean of 16 consecutive tokens. grid = B*256, block = 64.
__global__ void landmark_kernel(const float* __restrict__ q,
                                const float* __restrict__ k,
                                float* __restrict__ ql,
                                float* __restrict__ kl) {
  const int blk = blockIdx.x;
  const int d = threadIdx.x;
  size_t base = (size_t)blk * 16 * 64 + d;
  float sq = 0.f, sk = 0.f;
  for (int j = 0; j < 16; ++j) {
    sq += q[base + (size_t)j * 64];
    sk += k[base + (size_t)j * 64];
  }
  ql[(size_t)blk * 64 + d] = sq * (1.f / 16.f);
  kl[(size_t)blk * 64 + d] = sk * (1.f / 16.f);
}

// row softmax in place; one block (256 thr) per row.
__global__ void softmax_kernel(float* __restrict__ C, int cols) {
  float* p = C + (size_t)blockIdx.x * cols;
  __shared__ float red[256];
  const int t = threadIdx.x;
  float mx = -INFINITY;
  for (int c = t; c < cols; c += 256) mx = fmaxf(mx, p[c]);
  red[t] = mx;
  __syncthreads();
  for (int s = 128; s > 0; s >>= 1) {
    if (t < s) red[t] = fmaxf(red[t], red[t + s]);
    __syncthreads();
  }
  mx = red[0];
  __syncthreads();
  float sum = 0.f;
  for (int c = t; c < cols; c += 256) {
    float e = expf(p[c] - mx);
    p[c] = e;
    sum += e;
  }
  red[t] = sum;
  __syncthreads();
  for (int s = 128; s > 0; s >>= 1) {
    if (t < s) red[t] += red[t + s];
    __syncthreads();
  }
  float inv = 1.f / red[0];
  for (int c = t; c < cols; c += 256) p[c] *= inv;
}

// abs row-sum & col-sum of a2 (256x256 per batch). grid = B*256, block = 256.
__global__ void rowcol_kernel(const float* __restrict__ a2,
                              float* __restrict__ rs, float* __restrict__ cs) {
  const int idx = blockIdx.x;
  const int b = idx >> 8, r = idx & 255;
  const float* base = a2 + (size_t)b * 256 * 256;
  const int t = threadIdx.x;
  __shared__ float rr[256], cc[256];
  rr[t] = fabsf(base[(size_t)r * 256 + t]);
  cc[t] = fabsf(base[(size_t)t * 256 + r]);
  __syncthreads();
  for (int s = 128; s > 0; s >>= 1) {
    if (t < s) {
      rr[t] += rr[t + s];
      cc[t] += cc[t + s];
    }
    __syncthreads();
  }
  if (t == 0) {
    rs[idx] = rr[0];
    cs[idx] = cc[0];
  }
}

// scale = 1 / (max(rowsum) * max(colsum)) over all b,m. single block.
__global__ void scale_kernel(const float* __restrict__ rs,
                             const float* __restrict__ cs, int n,
                             float* __restrict__ out) {
  __shared__ float r1[256], c1[256];
  const int t = threadIdx.x;
  float mr = -INFINITY, mc = -INFINITY;
  for (int i = t; i < n; i += 256) {
    mr = fmaxf(mr, rs[i]);
    mc = fmaxf(mc, cs[i]);
  }
  r1[t] = mr;
  c1[t] = mc;
  __syncthreads();
  for (int s = 128; s > 0; s >>= 1) {
    if (t < s) {
      r1[t] = fmaxf(r1[t], r1[t + s]);
      c1[t] = fmaxf(c1[t], c1[t + s]);
    }
    __syncthreads();
  }
  if (t == 0) out[0] = 1.f / (r1[0] * c1[0]);
}

// z0[b][i][j] = a2[b][j][i] * scale
__global__ void zinit_kernel(float* __restrict__ z, const float* __restrict__ a2,
                             const float* __restrict__ scale) {
  const int b = blockIdx.z;
  const int i = blockIdx.y * 16 + threadIdx.y;
  const int j = blockIdx.x * 16 + threadIdx.x;
  z[((size_t)b * 256 + i) * 256 + j] =
      a2[((size_t)b * 256 + j) * 256 + i] * scale[0];
}

// dst = coef*I - src  (256x256 per batch). grid = B*256, block = 256.
__global__ void negI_kernel(float* __restrict__ dst, const float* __restrict__ src,
                            float coef) {
  const int row = blockIdx.x;
  const int i = row & 255;
  const int j = threadIdx.x;
  size_t idx = (size_t)row * 256 + j;
  dst[idx] = (i == j ? coef : 0.f) - src[idx];
}

// depthwise conv over seq dim (33 taps, zero pad), accumulate into outh.
__global__ void conv_add_kernel(float* __restrict__ outh,
                                const float* __restrict__ v,
                                const float* __restrict__ wc) {
  const int row = blockIdx.x;
  const int b = row >> 12, n = row & 4095;
  const int d = threadIdx.x;
  float s = 0.f;
#pragma unroll
  for (int kk = 0; kk < 33; ++kk) {
    int nn = n + kk - 16;
    if (nn >= 0 && nn < 4096)
      s += v[(((size_t)b << 12) + nn) * 64 + d] * wc[kk];
  }
  outh[(size_t)row * 64 + d] += s;
}

// ---------------------------------------------------------------------------
extern "C" void kernel_launch(void* const* d_in, const int* in_sizes, int n_in,
                              void* d_out, int out_size, void* d_ws,
                              size_t ws_size, hipStream_t stream) {
  const int Bn = 4, Nn = 4096, Dh = 64, LM = 256, DIMc = 512;
  const float* x = (const float*)d_in[0];
  const float* gamma = (const float*)d_in[1];
  const float* beta = (const float*)d_in[2];
  const float* wqkv = (const float*)d_in[3];
  const float* wout = (const float*)d_in[4];
  const float* bout = (const float*)d_in[5];
  const float* wconv = (const float*)d_in[6];
  float* out0 = (float*)d_out;                       // [B,N,512]
  float* attn = out0 + (size_t)Bn * Nn * DIMc;       // [B,N,N]

  float* ws = (float*)d_ws;
  size_t off = 0;
  auto alloc = [&](size_t n) { float* p = ws + off; off += n; return p; };
  float* q = alloc((size_t)Bn * Nn * Dh);
  float* k_ = alloc((size_t)Bn * Nn * Dh);
  float* v = alloc((size_t)Bn * Nn * Dh);
  float* ql = alloc((size_t)Bn * LM * Dh);
  float* kl = alloc((size_t)Bn * LM * Dh);
  float* a1 = alloc((size_t)Bn * Nn * LM);
  float* a2 = alloc((size_t)Bn * LM * LM);
  float* a3 = alloc((size_t)Bn * LM * Nn);
  float* z0 = alloc((size_t)Bn * LM * LM);
  float* z1 = alloc((size_t)Bn * LM * LM);
  float* t0 = alloc((size_t)Bn * LM * LM);
  float* t1 = alloc((size_t)Bn * LM * LM);
  float* t2 = alloc((size_t)Bn * LM * LM);
  float* a3v = alloc((size_t)Bn * LM * Dh);
  float* left = alloc((size_t)Bn * Nn * LM);
  float* outh = alloc((size_t)Bn * Nn * Dh);
  float* rsum = alloc(Bn * LM);
  float* csum = alloc(Bn * LM);
  float* scal = alloc(1);

  auto gemm = [&](int tB, float* C, const float* A, const float* Bp, int Mi,
                  int Ni, int Ki, int lda, int ldb, int ldc, long long sA,
                  long long sB, long long sC, int batches, float alpha,
                  const float* bias, const float* resid, int ldres) {
    dim3 g(Mi / 16, (Ni + 127) / 128, batches);
    if (tB)
      gemm_kernel<1><<<g, 256, 0, stream>>>(C, A, Bp, Mi, Ni, Ki, lda, ldb,
                                            ldc, sA, sB, sC, alpha, bias,
                                            resid, ldres);
    else
      gemm_kernel<0><<<g, 256, 0, stream>>>(C, A, Bp, Mi, Ni, Ki, lda, ldb,
                                            ldc, sA, sB, sC, alpha, bias,
                                            resid, ldres);
  };

  const long long sQKV = (long long)Nn * Dh;   // per-batch q/k/v stride
  const long long sL = (long long)LM * Dh;     // per-batch landmark stride
  const long long sMM = (long long)LM * LM;    // 256x256 stride
  const long long sNM = (long long)Nn * LM;
  const long long sMN = (long long)LM * Nn;

  // 1) LayerNorm + QKV
  ln_qkv_kernel<<<Bn * Nn / 16, 128, 0, stream>>>(x, gamma, beta, wqkv, q, k_, v);
  // 2) landmarks
  landmark_kernel<<<Bn * LM, 64, 0, stream>>>(q, k_, ql, kl);
  // 3) sims (NT, K=64)
  gemm(1, a1, q, kl, Nn, LM, 64, 64, 64, LM, sQKV, sL, sNM, Bn, 1.f, nullptr, nullptr, 0);
  gemm(1, a2, ql, kl, LM, LM, 64, 64, 64, LM, sL, sL, sMM, Bn, 1.f, nullptr, nullptr, 0);
  gemm(1, a3, ql, k_, LM, Nn, 64, 64, 64, Nn, sL, sQKV, sMN, Bn, 1.f, nullptr, nullptr, 0);
  // 4) softmaxes
  softmax_kernel<<<Bn * Nn, 256, 0, stream>>>(a1, LM);
  softmax_kernel<<<Bn * LM, 256, 0, stream>>>(a2, LM);
  softmax_kernel<<<Bn * LM, 256, 0, stream>>>(a3, Nn);
  // 5) pinv init: z = a2^T / (max(rowsum)*max(colsum))
  rowcol_kernel<<<Bn * LM, 256, 0, stream>>>(a2, rsum, csum);
  scale_kernel<<<1, 256, 0, stream>>>(rsum, csum, Bn * LM, scal);
  zinit_kernel<<<dim3(LM / 16, LM / 16, Bn), dim3(16, 16), 0, stream>>>(z0, a2, scal);
  // 6) Newton-Schulz iterations
  float* zc = z0;
  float* za = z1;
  for (int it = 0; it < 6; ++it) {
    gemm(0, t0, a2, zc, LM, LM, LM, LM, LM, LM, sMM, sMM, sMM, Bn, 1.f, nullptr, nullptr, 0);
    negI_kernel<<<Bn * LM, 256, 0, stream>>>(t1, t0, 7.f);
    gemm(0, t2, t0, t1, LM, LM, LM, LM, LM, LM, sMM, sMM, sMM, Bn, 1.f, nullptr, nullptr, 0);
    negI_kernel<<<Bn * LM, 256, 0, stream>>>(t1, t2, 15.f);
    gemm(0, t2, t0, t1, LM, LM, LM, LM, LM, LM, sMM, sMM, sMM, Bn, 1.f, nullptr, nullptr, 0);
    negI_kernel<<<Bn * LM, 256, 0, stream>>>(t1, t2, 13.f);
    gemm(0, za, zc, t1, LM, LM, LM, LM, LM, LM, sMM, sMM, sMM, Bn, 0.25f, nullptr, nullptr, 0);
    float* tmp = zc; zc = za; za = tmp;
  }
  // 7) a3v = a3 @ v   [256,64] K=4096
  gemm(0, a3v, a3, v, LM, Dh, Nn, Nn, Dh, Dh, sMN, sQKV, sL, Bn, 1.f, nullptr, nullptr, 0);
  // 8) left = a1 @ z  [4096,256] K=256
  gemm(0, left, a1, zc, Nn, LM, LM, LM, LM, LM, sNM, sMM, sNM, Bn, 1.f, nullptr, nullptr, 0);
  // 9) outh = left @ a3v  [4096,64] K=256
  gemm(0, outh, left, a3v, Nn, Dh, LM, LM, Dh, Dh, sNM, sL, sQKV, Bn, 1.f, nullptr, nullptr, 0);
  // 10) conv residual into outh
  conv_add_kernel<<<Bn * Nn, 64, 0, stream>>>(outh, v, wconv);
  // 11) y = x + outh @ w_out + b_out  (fold batch into M)
  gemm(0, out0, outh, wout, Bn * Nn, DIMc, Dh, Dh, DIMc, DIMc, 0, 0, 0, 1, 1.f,
       bout, x, DIMc);
  // 12) attn = left @ a3  [4096,4096] K=256
  gemm(0, attn, left, a3, Nn, Nn, LM, LM, Nn, Nn, sNM, sMN, (long long)Nn * Nn,
       Bn, 1.f, nullptr, nullptr, 0);
}